// Model_10488310137417
// MI455X (gfx1250) — compile-verified
//
#include <hip/hip_runtime.h>
#include <hip/hip_bf16.h>

typedef float v2f __attribute__((ext_vector_type(2)));
typedef float v8f __attribute__((ext_vector_type(8)));

#define KM_K 128
#define KM_N 128
#define KM_TOL2 1e-8f   // tol=1e-4, compare squared norms

// ---------------------------------------------------------------------------
// init: cent = cent0, sums = 0, counts = 0, flags = {done=0, any_ge=0}
// ---------------------------------------------------------------------------
__global__ void kmeans_init(const float* __restrict__ cent0,
                            float* __restrict__ cent,
                            float* __restrict__ sums,
                            float* __restrict__ counts,
                            int* __restrict__ flags) {
    int i = blockIdx.x * 256 + threadIdx.x;
    if (i < KM_K * KM_N) {
        cent[i] = cent0[i];
        sums[i] = 0.0f;
    }
    if (i < KM_K) counts[i] = 0.0f;
    if (i < 2) flags[i] = 0;
}

// ---------------------------------------------------------------------------
// assign: d2' = c_sq - 2 * (X @ C^T), argmin over 128 clusters.
// 256 threads = 8 waves; wave w handles rows [blk*128 + 16w, +16).
// Uses V_WMMA_F32_16X16X4_F32, K-loop of 32 steps over the 128-dim.
// ---------------------------------------------------------------------------
__global__ __launch_bounds__(256) void kmeans_assign(
    const float* __restrict__ X, const float* __restrict__ cent,
    const float* __restrict__ c_sq, const int* __restrict__ flags,
    float* __restrict__ clusters_out, int m) {
    if (flags[0]) return;  // converged earlier: keep previous clusters

    __shared__ float sC[KM_K * KM_N];  // centroids, row-major
    __shared__ float sQ[KM_K];         // c_sq

    const int t = threadIdx.x;
    // Stage centroids into LDS with 128-bit loads
    {
        const float4* g4 = (const float4*)cent;
        float4* s4 = (float4*)sC;
        #pragma unroll
        for (int i = 0; i < (KM_K * KM_N / 4) / 256; ++i)
            s4[t + i * 256] = g4[t + i * 256];
        if (t < KM_K) sQ[t] = c_sq[t];
    }
    __syncthreads();

    const int wave = t >> 5;        // 0..7
    const int lane = t & 31;
    const int half = lane >> 4;     // 0 or 1
    const int lr   = lane & 15;
    const long rowBase = (long)blockIdx.x * 128 + wave * 16;
    long row = rowBase + lr;
    if (row >= m) row = 0;          // clamp; garbage results never used
    const float* __restrict__ xrow = X + (size_t)row * KM_N;

    v8f acc[8];
    #pragma unroll
    for (int ti = 0; ti < 8; ++ti) acc[ti] = (v8f){0,0,0,0,0,0,0,0};

    // A-frag (16x4 f32): lane lr holds M=lr, VGPR j holds K = j + 2*half.
    // B-frag (4x16 f32): lane lr holds N=lr, VGPR j holds K = j + 2*half
    //   -> B[k][n] = C[n][k], i.e. float2 from centroid row (16*ti + lr).
    for (int k = 0; k < KM_N; k += 4) {
        float2 af = *(const float2*)(xrow + k + 2 * half);
        v2f a; a.x = af.x; a.y = af.y;
        #pragma unroll
        for (int ti = 0; ti < 8; ++ti) {
            const float* cr = sC + (size_t)(16 * ti + lr) * KM_N + k + 2 * half;
            float2 bf = *(const float2*)cr;
            v2f b; b.x = bf.x; b.y = bf.y;
            acc[ti] = __builtin_amdgcn_wmma_f32_16x16x4_f32(
                false, a, false, b, (short)0, acc[ti], false, false);
        }
    }

    // D layout: VGPR j of lane -> row (rowBase + j + 8*half), col (16*ti + lr)
    float minv[8];
    int   mini[8];
    #pragma unroll
    for (int j = 0; j < 8; ++j) { minv[j] = 3.4e38f; mini[j] = 0; }
    #pragma unroll
    for (int ti = 0; ti < 8; ++ti) {
        const int col = 16 * ti + lr;
        const float q = sQ[col];
        #pragma unroll
        for (int j = 0; j < 8; ++j) {
            float v = q - 2.0f * acc[ti][j];
            if (v < minv[j]) { minv[j] = v; mini[j] = col; }  // cols increase with ti
        }
    }
    // Cross-lane argmin over the 16 lanes of each half (first-index ties)
    #pragma unroll
    for (int off = 1; off < 16; off <<= 1) {
        #pragma unroll
        for (int j = 0; j < 8; ++j) {
            float ov = __shfl_xor(minv[j], off, 32);
            int   oi = __shfl_xor(mini[j], off, 32);
            if (ov < minv[j] || (ov == minv[j] && oi < mini[j])) {
                minv[j] = ov; mini[j] = oi;
            }
        }
    }
    if (lr == 0) {
        #pragma unroll
        for (int j = 0; j < 8; ++j) {
            long r = rowBase + j + 8 * half;
            if (r < m) clusters_out[r] = (float)mini[j];
        }
    }
}

// ---------------------------------------------------------------------------
// accum: segment-sum of X by cluster id. Each WG owns 1024 points,
// accumulates into a 64KB LDS tile, flushes once with global f32 atomics.
// 256 threads: two points at a time, 128 dims per point.
// ---------------------------------------------------------------------------
__global__ __launch_bounds__(256) void kmeans_accum(
    const float* __restrict__ X, const float* __restrict__ clusters,
    const int* __restrict__ flags, float* __restrict__ sums,
    float* __restrict__ counts, int m) {
    if (flags[0]) return;

    __shared__ float sS[KM_K * KM_N];
    __shared__ float sCnt[KM_K];
    const int t = threadIdx.x;
    for (int i = t; i < KM_K * KM_N; i += 256) sS[i] = 0.0f;
    if (t < KM_K) sCnt[t] = 0.0f;
    __syncthreads();

    const int d = t & 127;
    const int sub = t >> 7;  // 0 or 1
    const long base = (long)blockIdx.x * 1024;
    for (int i = 0; i < 1024; i += 2) {
        const long r = base + i + sub;
        if (r < m) {
            const int c = (int)clusters[r];
            const float v = X[(size_t)r * KM_N + d];
            atomicAdd(&sS[c * KM_N + d], v);
            if (d == 0) atomicAdd(&sCnt[c], 1.0f);
        }
    }
    __syncthreads();
    for (int i = t; i < KM_K * KM_N; i += 256) atomicAdd(&sums[i], sS[i]);
    if (t < KM_K) atomicAdd(&counts[t], sCnt[t]);
}

// ---------------------------------------------------------------------------
// update1: per-cluster new centroid + convergence test; zero sums/counts.
// One WG (128 threads) per cluster.
// ---------------------------------------------------------------------------
__global__ __launch_bounds__(128) void kmeans_update1(
    const float* __restrict__ cent, float* __restrict__ cent_new,
    float* __restrict__ sums, float* __restrict__ counts,
    int* __restrict__ flags) {
    const int k = blockIdx.x, d = threadIdx.x;
    __shared__ float red[128];
    const float cnt = counts[k];
    const float s = sums[k * KM_N + d];
    const float old = cent[k * KM_N + d];
    const float nv = (cnt > 0.0f) ? (s / fmaxf(cnt, 1.0f)) : old;
    cent_new[k * KM_N + d] = nv;
    sums[k * KM_N + d] = 0.0f;  // ready for next iteration
    const float df = nv - old;
    red[d] = df * df;
    __syncthreads();
    for (int off = 64; off > 0; off >>= 1) {
        if (d < off) red[d] += red[d + off];
        __syncthreads();
    }
    if (d == 0) {
        counts[k] = 0.0f;
        if (red[0] >= KM_TOL2) atomicOr(&flags[1], 1);  // some cluster moved >= tol
    }
}

// ---------------------------------------------------------------------------
// update2: commit cent = cent_new iff (!done && !converged); refresh c_sq.
// Also used once after init (flags all 0 -> just computes c_sq of cent).
// ---------------------------------------------------------------------------
__global__ __launch_bounds__(128) void kmeans_update2(
    float* __restrict__ cent, const float* __restrict__ cent_new,
    float* __restrict__ c_sq, const int* __restrict__ flags) {
    const int k = blockIdx.x, d = threadIdx.x;
    __shared__ float red[128];
    const bool upd = (flags[0] == 0) && (flags[1] != 0);
    float v = upd ? cent_new[k * KM_N + d] : cent[k * KM_N + d];
    if (upd) cent[k * KM_N + d] = v;
    red[d] = v * v;
    __syncthreads();
    for (int off = 64; off > 0; off >>= 1) {
        if (d < off) red[d] += red[d + off];
        __syncthreads();
    }
    if (d == 0) c_sq[k] = red[0];
}

// update3: done |= converged; reset any_ge.
__global__ void kmeans_update3(int* __restrict__ flags) {
    if (flags[1] == 0) flags[0] = 1;
    flags[1] = 0;
}

// final: centroids -> d_out[0:16384]
__global__ void kmeans_final(const float* __restrict__ cent,
                             float* __restrict__ out) {
    int i = blockIdx.x * 256 + threadIdx.x;
    if (i < KM_K * KM_N) out[i] = cent[i];
}

extern "C" void kernel_launch(void* const* d_in, const int* in_sizes, int n_in,
                              void* d_out, int out_size, void* d_ws, size_t ws_size,
                              hipStream_t stream) {
    const float* X  = (const float*)d_in[0];
    const float* C0 = (const float*)d_in[1];
    const int m = in_sizes[0] / KM_N;  // 100000

    float* out = (float*)d_out;
    float* clusters = out + KM_K * KM_N;  // clusters written as float indices

    float* cent     = (float*)d_ws;
    float* cent_new = cent + KM_K * KM_N;
    float* sums     = cent_new + KM_K * KM_N;
    float* counts   = sums + KM_K * KM_N;
    float* c_sq     = counts + KM_K;
    int*   flags    = (int*)(c_sq + KM_K);  // [0]=done, [1]=any_ge

    kmeans_init<<<64, 256, 0, stream>>>(C0, cent, sums, counts, flags);
    kmeans_update2<<<KM_K, 128, 0, stream>>>(cent, cent_new, c_sq, flags); // c_sq init

    const int gridA = (m + 127) / 128;
    const int gridS = (m + 1023) / 1024;
    for (int it = 0; it < 15; ++it) {
        kmeans_assign<<<gridA, 256, 0, stream>>>(X, cent, c_sq, flags, clusters, m);
        kmeans_accum<<<gridS, 256, 0, stream>>>(X, clusters, flags, sums, counts, m);
        kmeans_update1<<<KM_K, 128, 0, stream>>>(cent, cent_new, sums, counts, flags);
        kmeans_update2<<<KM_K, 128, 0, stream>>>(cent, cent_new, c_sq, flags);
        kmeans_update3<<<1, 1, 0, stream>>>(flags);
    }
    kmeans_final<<<64, 256, 0, stream>>>(cent, out);
}